// GNN_11149735101019
// MI455X (gfx1250) — compile-verified
//
#include <hip/hip_runtime.h>
#include <hip/hip_bf16.h>

// ---------------------------------------------------------------------------
// GraphConv x3 + mean-pool + linear for MI455X (gfx1250, wave32, WMMA).
// Edge scatter is HBM/L2 bound; dense 64x64 GEMMs use V_WMMA_F32_16X16X4_F32
// to keep fp32 precision (GEMM is negligible vs. scatter traffic).
// ---------------------------------------------------------------------------

#define GNN_N_NODES  100000
#define GNN_N_EDGES  1600000
#define GNN_HID      64
#define GNN_OUT      16
#define GNN_GRAPHS   64

typedef float v2f __attribute__((ext_vector_type(2)));
typedef float v8f __attribute__((ext_vector_type(8)));

// ----------------------------- utility kernels -----------------------------

__global__ void gnn_zero_f32(float* __restrict__ p, int n) {
    int i = blockIdx.x * blockDim.x + threadIdx.x;
    if (i < n) p[i] = 0.0f;
}

// Layer-1 edge aggregation: IN_CH == 1, scalar per edge.
__global__ void gnn_edge_agg_scalar(const float* __restrict__ x,
                                    const int* __restrict__ ei,
                                    const float* __restrict__ w,
                                    float* __restrict__ agg1) {
    int e = blockIdx.x * blockDim.x + threadIdx.x;
    if (e >= GNN_N_EDGES) return;
    int s = ei[e];
    int d = ei[GNN_N_EDGES + e];
    atomicAdd(&agg1[d], x[s] * w[e]);
}

// Layer-1 expansion: h[n,c] = relu(agg1[n]*W_rel1[c] + b1[c] + x[n]*W_root1[c])
__global__ void gnn_layer1_expand(const float* __restrict__ agg1,
                                  const float* __restrict__ x,
                                  const float* __restrict__ Wr,
                                  const float* __restrict__ br,
                                  const float* __restrict__ Wo,
                                  float* __restrict__ h) {
    int i = blockIdx.x * blockDim.x + threadIdx.x;
    if (i >= GNN_N_NODES * GNN_HID) return;
    int n = i >> 6;
    int c = i & 63;
    float v = agg1[n] * Wr[c] + br[c] + x[n] * Wo[c];
    h[i] = fmaxf(v, 0.0f);
}

// Vector edge aggregation: one thread per (edge, channel). Coalesced gather
// from h[src] and coalesced atomic scatter into agg[dst].
__global__ void gnn_edge_agg_vec(const float* __restrict__ h,
                                 const int* __restrict__ ei,
                                 const float* __restrict__ w,
                                 float* __restrict__ agg) {
    int i = blockIdx.x * blockDim.x + threadIdx.x;   // < E*64 = 102.4M, fits int
    if (i >= GNN_N_EDGES * GNN_HID) return;
    int e = i >> 6;
    int c = i & 63;
    int s = ei[e];
    int d = ei[GNN_N_EDGES + e];
    atomicAdd(&agg[d * GNN_HID + c], h[s * GNN_HID + c] * w[e]);
}

// ----------------------- WMMA dense layer (fp32 GEMM) -----------------------
// out[n,:] = agg[n,:] @ Wrel.T + hin[n,:] @ Wroot.T + bias  (optional ReLU)
// 4 waves/block, each wave owns one 16-row node tile and all 64 output cols
// (4 column tiles of 16). K=64 -> 16 steps of V_WMMA_F32_16X16X4_F32 per
// (tile, matrix) pair. Weights (2 x 16KB) staged in LDS.
//
// ISA layouts (wave32, 32-bit data):
//   A 16x4 : lane L holds A[L%16, 2*(L/16)+v] in vgpr v (v = 0,1)
//   B 4x16 : lane L holds B[2*(L/16)+v, L%16] in vgpr v
//   C/D    : vgpr r, lanes 0-15 -> (M=r, N=lane); lanes 16-31 -> (M=r+8)
// `out` may alias `agg` (in-place): each wave reads its 16 rows fully in the
// K-loop before the epilogue stores, and no other wave touches those rows.
template <bool RELU>
__global__ __launch_bounds__(128, 2) void gnn_dense_wmma(
        const float* agg, const float* hin,
        const float* __restrict__ Wrel, const float* __restrict__ Wroot,
        const float* __restrict__ bias, float* out, int nTiles) {
    __shared__ float sW[2 * GNN_HID * GNN_HID];     // [0..4095]=Wrel, [4096..]=Wroot
    for (int i = (int)threadIdx.x; i < 2 * GNN_HID * GNN_HID; i += 128)
        sW[i] = (i < GNN_HID * GNN_HID) ? Wrel[i] : Wroot[i - GNN_HID * GNN_HID];
    __syncthreads();

    const int wave = threadIdx.x >> 5;
    const int lane = threadIdx.x & 31;
    const int tile = blockIdx.x * 4 + wave;
    if (tile >= nTiles) return;                     // uniform per wave: EXEC all-1s

    const int lm = lane & 15;                       // M (A) / N (B, C, D)
    const int kh = (lane >> 4) << 1;                // K sub-offset: 0 or 2

    const float* aRow = agg + (size_t)(tile * 16 + lm) * GNN_HID;
    const float* hRow = hin + (size_t)(tile * 16 + lm) * GNN_HID;

    v8f acc[4];
#pragma unroll
    for (int c = 0; c < 4; ++c) { v8f z = {}; acc[c] = z; }

#pragma unroll
    for (int k0 = 0; k0 < GNN_HID; k0 += 4) {
        const int ka = k0 + kh;
        v2f aRel;  aRel.x  = aRow[ka]; aRel.y  = aRow[ka + 1];
        v2f aRoot; aRoot.x = hRow[ka]; aRoot.y = hRow[ka + 1];
#pragma unroll
        for (int c = 0; c < 4; ++c) {
            const int n = c * 16 + lm;              // output channel
            // B[k][n] = W[n][k]  (out = A @ W.T, W row-major [64 out][64 in])
            v2f bRel;  bRel.x  = sW[n * GNN_HID + ka];
                       bRel.y  = sW[n * GNN_HID + ka + 1];
            v2f bRoot; bRoot.x = sW[GNN_HID * GNN_HID + n * GNN_HID + ka];
                       bRoot.y = sW[GNN_HID * GNN_HID + n * GNN_HID + ka + 1];
            acc[c] = __builtin_amdgcn_wmma_f32_16x16x4_f32(
                false, aRel, false, bRel, (short)0, acc[c], false, false);
            acc[c] = __builtin_amdgcn_wmma_f32_16x16x4_f32(
                false, aRoot, false, bRoot, (short)0, acc[c], false, false);
        }
    }

    const int mbase = tile * 16 + ((lane >> 4) << 3);   // upper lane half -> +8
#pragma unroll
    for (int c = 0; c < 4; ++c) {
        const float bv = bias[c * 16 + lm];
#pragma unroll
        for (int r = 0; r < 8; ++r) {
            float v = acc[c][r] + bv;
            if (RELU) v = fmaxf(v, 0.0f);
            out[(size_t)(mbase + r) * GNN_HID + c * 16 + lm] = v;
        }
    }
}

// ------------------------------ pooling / head ------------------------------

__global__ void gnn_pool(const float* __restrict__ h,
                         const int* __restrict__ batch,
                         float* __restrict__ pool,
                         float* __restrict__ cnt) {
    int i = blockIdx.x * blockDim.x + threadIdx.x;
    if (i >= GNN_N_NODES * GNN_HID) return;
    int n = i >> 6;
    int c = i & 63;
    int g = batch[n];
    atomicAdd(&pool[g * GNN_HID + c], h[i]);
    if (c == 0) atomicAdd(&cnt[g], 1.0f);
}

__global__ void gnn_final_linear(const float* __restrict__ pool,
                                 const float* __restrict__ cnt,
                                 const float* __restrict__ Wl,
                                 const float* __restrict__ bl,
                                 float* __restrict__ out) {
    int i = blockIdx.x * blockDim.x + threadIdx.x;
    if (i >= GNN_GRAPHS * GNN_OUT) return;
    int g = i >> 4;
    int o = i & 15;
    float inv = 1.0f / fmaxf(cnt[g], 1.0f);
    float s = bl[o];
#pragma unroll 8
    for (int k = 0; k < GNN_HID; ++k)
        s += pool[g * GNN_HID + k] * inv * Wl[o * GNN_HID + k];
    out[i] = s;
}

// --------------------------------- launcher ---------------------------------

extern "C" void kernel_launch(void* const* d_in, const int* in_sizes, int n_in,
                              void* d_out, int out_size, void* d_ws, size_t ws_size,
                              hipStream_t stream) {
    (void)in_sizes; (void)n_in; (void)out_size; (void)ws_size;

    const float* x     = (const float*)d_in[0];
    const int*   ei    = (const int*)  d_in[1];   // [2, E]
    const float* ew    = (const float*)d_in[2];
    const int*   batch = (const int*)  d_in[3];
    const float* Wr1   = (const float*)d_in[4];
    const float* br1   = (const float*)d_in[5];
    const float* Wo1   = (const float*)d_in[6];
    const float* Wr2   = (const float*)d_in[7];
    const float* br2   = (const float*)d_in[8];
    const float* Wo2   = (const float*)d_in[9];
    const float* Wr3   = (const float*)d_in[10];
    const float* br3   = (const float*)d_in[11];
    const float* Wo3   = (const float*)d_in[12];
    const float* Wl    = (const float*)d_in[13];
    const float* bl    = (const float*)d_in[14];
    float* out = (float*)d_out;

    // Workspace: two ping-pong N x 64 buffers + pool sums + counts.
    const size_t NH = (size_t)GNN_N_NODES * GNN_HID;     // 6.4M floats
    float* bufA = (float*)d_ws;          // h(t)
    float* bufB = bufA + NH;             // agg / h(t+1) (dense writes in place)
    float* pool = bufB + NH;             // 64*64
    float* cnt  = pool + GNN_GRAPHS * GNN_HID;           // 64

    const int TPB = 256;
    const int nhBlocks   = (int)((NH + TPB - 1) / TPB);                  // 25000
    const int edgeBlocks = (GNN_N_EDGES + TPB - 1) / TPB;                // 6250
    const int evBlocks   = (GNN_N_EDGES * GNN_HID + TPB - 1) / TPB;      // 400000
    const int nTiles     = GNN_N_NODES / 16;                             // 6250
    const int denseBlocks = (nTiles + 3) / 4;                            // 1563

    // ---- Layer 1 (IN_CH = 1): scalar scatter + outer-product expansion ----
    gnn_zero_f32<<<(GNN_N_NODES + TPB - 1) / TPB, TPB, 0, stream>>>(bufB, GNN_N_NODES);
    gnn_edge_agg_scalar<<<edgeBlocks, TPB, 0, stream>>>(x, ei, ew, bufB);
    gnn_layer1_expand<<<nhBlocks, TPB, 0, stream>>>(bufB, x, Wr1, br1, Wo1, bufA);

    // ---- Layer 2: vector scatter into bufB, WMMA dense (ReLU) in-place ----
    gnn_zero_f32<<<nhBlocks, TPB, 0, stream>>>(bufB, (int)NH);
    gnn_edge_agg_vec<<<evBlocks, TPB, 0, stream>>>(bufA, ei, ew, bufB);
    gnn_dense_wmma<true><<<denseBlocks, 128, 0, stream>>>(bufB, bufA, Wr2, Wo2, br2,
                                                          bufB, nTiles);   // h2 = bufB

    // ---- Layer 3: vector scatter into bufA, WMMA dense (no ReLU) ----------
    gnn_zero_f32<<<nhBlocks, TPB, 0, stream>>>(bufA, (int)NH);
    gnn_edge_agg_vec<<<evBlocks, TPB, 0, stream>>>(bufB, ei, ew, bufA);
    gnn_dense_wmma<false><<<denseBlocks, 128, 0, stream>>>(bufA, bufB, Wr3, Wo3, br3,
                                                           bufA, nTiles);  // h3 = bufA

    // ---- Mean pool + final linear -----------------------------------------
    gnn_zero_f32<<<(GNN_GRAPHS * GNN_HID + GNN_GRAPHS + TPB - 1) / TPB, TPB, 0, stream>>>(
        pool, GNN_GRAPHS * GNN_HID + GNN_GRAPHS);
    gnn_pool<<<nhBlocks, TPB, 0, stream>>>(bufA, batch, pool, cnt);
    gnn_final_linear<<<(GNN_GRAPHS * GNN_OUT + TPB - 1) / TPB, TPB, 0, stream>>>(
        pool, cnt, Wl, bl, out);
}